// VoxelSampler_traj_6657199309192
// MI455X (gfx1250) — compile-verified
//
#include <hip/hip_runtime.h>

// Problem constants from the reference.
#define B_ 2
#define T_ 8
#define N_ 65536
#define M_ 128
#define F_ 5
#define S_ 128            // num_sample
#define GAMMA_ 1.1f

typedef __attribute__((ext_vector_type(2))) float v2f;
typedef __attribute__((ext_vector_type(8))) float v8f;

// One wave (32 lanes) per group of 16 ROIs of one (b, t) frame.
// Distance matrix computed as a rank-4 GEMM on the WMMA unit:
//   d^2[m][n] = A[m] . B[:,n],  A[m] = (-2cx, -2cy, cx^2+cy^2, 1),
//                               B[:,n] = (px, py, 1, px^2+py^2)
// 32 points per iteration via two WMMAs sharing the A operand; point loads
// are software-pipelined one iteration ahead to hide VMEM latency behind
// the XDL + compaction work.
__global__ __launch_bounds__(32)
void vs_scan_kernel(const float* __restrict__ points,
                    const float* __restrict__ rois,
                    float* __restrict__ out) {
    __shared__ unsigned int idxbuf[16 * S_];   // first-128 point indices per ROI
    __shared__ int cnts[16];

    const int lane = threadIdx.x;              // 0..31
    const int blk  = blockIdx.x;               // b*(T_*8) + t*8 + mg
    const int mg   = blk & 7;
    const int t    = (blk >> 3) & 7;
    const int b    = blk >> 6;
    const int m0   = mg * 16;

    const float* pts = points + (size_t)(b * T_ + t) * N_ * F_;

    // Each half-wave redundantly loads the box for ROI m0 + (lane & 15).
    const int mr = lane & 15;
    const float* box = rois + ((size_t)(b * T_ + t) * M_ + (m0 + mr)) * 7;
    const float cx = box[0], cy = box[1];
    const float bl = box[3], bw = box[4];
    const float rsq = 0.25f * (bl * bl + bw * bw) * (GAMMA_ * GAMMA_); // (|[l/2,w/2]|*gamma)^2
    const float nrm = cx * cx + cy * cy;

    const bool lowHalf = (lane < 16);

    // A-matrix (16x4 f32): lanes 0-15 hold K=0 (VGPR0) / K=1 (VGPR1),
    // lanes 16-31 hold K=2 / K=3 for the same row M = lane & 15.
    v2f amat;
    amat.x = lowHalf ? (-2.0f * cx) : nrm;
    amat.y = lowHalf ? (-2.0f * cy) : 1.0f;

    // D-tile VGPR v holds row M=v (lanes 0-15) and M=v+8 (lanes 16-31):
    // per-lane radius^2 for each of the 8 accumulator registers.
    float rv[8];
#pragma unroll
    for (int v = 0; v < 8; ++v)
        rv[v] = __shfl(rsq, lowHalf ? v : (v + 8), 32);

    int cnt = lowHalf ? 0 : 0x7fffffff;        // high lanes are permanently "done"

    // Pipeline prologue: load the two 16-point tiles of iteration 0.
    float cpx0, cpy0, cpx1, cpy1;
    {
        const float* p0 = pts + (size_t)(mr) * F_;
        const float* p1 = pts + (size_t)(16 + mr) * F_;
        cpx0 = p0[0]; cpy0 = p0[1];
        cpx1 = p1[0]; cpy1 = p1[1];
    }

    for (int n0 = 0; n0 < N_; n0 += 32) {
        // Issue next iteration's point loads first (index-clamped; the clamped
        // values are never consumed on the final iteration).
        int i0 = n0 + 32 + mr;      if (i0 > N_ - 1) i0 = N_ - 1;
        int i1 = n0 + 48 + mr;      if (i1 > N_ - 1) i1 = N_ - 1;
        const float* q0 = pts + (size_t)i0 * F_;
        const float* q1 = pts + (size_t)i1 * F_;
        const float npx0 = q0[0], npy0 = q0[1];
        const float npx1 = q1[0], npy1 = q1[1];

        // Speculative prefetch further ahead (global_prefetch_b8; dropped if OOB).
        __builtin_prefetch(pts + (size_t)(n0 + 512 + mr) * F_, 0, 0);

        // B-matrices (4x16 f32): column n = point index; lanes 0-15 carry
        // (px, py) in K=0/K=1, lanes 16-31 carry (1, |p|^2) in K=2/K=3.
        v2f bmat0, bmat1;
        bmat0.x = lowHalf ? cpx0 : 1.0f;
        bmat0.y = lowHalf ? cpy0 : (cpx0 * cpx0 + cpy0 * cpy0);
        bmat1.x = lowHalf ? cpx1 : 1.0f;
        bmat1.y = lowHalf ? cpy1 : (cpx1 * cpx1 + cpy1 * cpy1);

        v8f acc0 = {0.f, 0.f, 0.f, 0.f, 0.f, 0.f, 0.f, 0.f};
        v8f acc1 = {0.f, 0.f, 0.f, 0.f, 0.f, 0.f, 0.f, 0.f};
        acc0 = __builtin_amdgcn_wmma_f32_16x16x4_f32(
            false, amat, false, bmat0, (short)0, acc0, false, false);
        acc1 = __builtin_amdgcn_wmma_f32_16x16x4_f32(
            false, amat, false, bmat1, (short)0, acc1, false, false);

        // One compare+ballot per D VGPR covers two ROIs (rows v and v+8)
        // per tile. Lane r (r<16) collects ROI r's masks for both tiles and
        // fuses them: bit j -> point n0+j, bit 16+j -> point n0+16+j.
        unsigned int mlo = 0u, mhi = 0u;
#pragma unroll
        for (int v = 0; v < 8; ++v) {
            const unsigned int ba = (unsigned int)__ballot(acc0[v] <= rv[v]);
            const unsigned int bb = (unsigned int)__ballot(acc1[v] <= rv[v]);
            if (lane == v)     { mlo = ba & 0xFFFFu; mhi = bb & 0xFFFFu; }
            if (lane == v + 8) { mlo = ba >> 16;     mhi = bb >> 16;     }
        }
        unsigned int mymask = mlo | (mhi << 16);

        // Append set bits LSB-first => ascending point index order, exactly
        // matching stable top_k on the 0/1 mask. Stop at S_ samples.
        while (mymask != 0u && cnt < S_) {
            const int bpos = __ffs(mymask) - 1;
            idxbuf[lane * S_ + cnt] = (unsigned int)(n0 + bpos);
            ++cnt;
            mymask &= (mymask - 1u);
        }

        // Rotate the pipelined loads.
        cpx0 = npx0; cpy0 = npy0; cpx1 = npx1; cpy1 = npy1;

        // Uniform early exit when all 16 ROIs have 128 samples.
        const unsigned int done = (unsigned int)__ballot(cnt >= S_);
        if (done == 0xFFFFFFFFu) break;
    }

    if (lowHalf) cnts[lane] = cnt;
    __syncthreads();

    // Emit: out[b, m, t*S + s, f]; zero-pad slots beyond the found count.
    for (int slot = lane; slot < 16 * S_; slot += 32) {
        const int r = slot >> 7;
        const int s = slot & (S_ - 1);
        const size_t obase = ((((size_t)b * M_ + (m0 + r)) * T_ + t) * S_ + s) * F_;
        if (s < cnts[r]) {
            const float* p = pts + (size_t)idxbuf[r * S_ + s] * F_;
#pragma unroll
            for (int f = 0; f < F_; ++f) out[obase + f] = p[f];
        } else {
#pragma unroll
            for (int f = 0; f < F_; ++f) out[obase + f] = 0.0f;
        }
    }
}

// Fallback: for t != 0 with valid_length == 0, the reference replaces the whole
// per-ROI sample block with the frame-0 block. Kernel 1 already wrote all of
// out, and t=0 slices are never overwritten here, so this is a safe d2d copy.
__global__ __launch_bounds__(128)
void vs_fallback_kernel(const int* __restrict__ valid,
                        float* __restrict__ out) {
    const int blk = blockIdx.x;                // b*T_*M_ + t*M_ + m
    const int m = blk % M_;
    const int t = (blk / M_) % T_;
    const int b = blk / (M_ * T_);
    if (t == 0) return;
    if (valid[((size_t)b * T_ + t) * M_ + m] != 0) return;
    const size_t dst = (((size_t)b * M_ + m) * T_ + t) * (size_t)(S_ * F_);
    const size_t src = (((size_t)b * M_ + m) * T_ + 0) * (size_t)(S_ * F_);
    for (int i = threadIdx.x; i < S_ * F_; i += 128)
        out[dst + i] = out[src + i];
}

extern "C" void kernel_launch(void* const* d_in, const int* in_sizes, int n_in,
                              void* d_out, int out_size, void* d_ws, size_t ws_size,
                              hipStream_t stream) {
    (void)in_sizes; (void)n_in; (void)d_ws; (void)ws_size; (void)out_size;
    const float* points = (const float*)d_in[0];   // [B,T,N,F] f32
    const float* rois   = (const float*)d_in[1];   // [B,T,M,7] f32
    const int*   valid  = (const int*)d_in[2];     // [B,T,M] i32
    // d_in[3] = num_sample (== 128, compile-time constant here)
    float* out = (float*)d_out;                    // [B,M,T*S,F] f32

    vs_scan_kernel<<<B_ * T_ * 8, 32, 0, stream>>>(points, rois, out);
    vs_fallback_kernel<<<B_ * T_ * M_, 128, 0, stream>>>(valid, out);
}